// AttCosine_90924457657000
// MI455X (gfx1250) — compile-verified
//
#include <hip/hip_runtime.h>

typedef __attribute__((ext_vector_type(2))) float v2f;
typedef __attribute__((ext_vector_type(8))) float v8f;

#define BATCH 8
#define LSEQ  2048
#define DDIM  64
#define EPSN  1e-8f

// ---------------------------------------------------------------------------
// Kernel 1: reciprocal L2 norms of every row of query and y.
// rows [0, 16384) -> query, rows [16384, 32768) -> y.  Output: d_ws float[32768].
// One wave per row: each lane loads a float2 (64 floats / 32 lanes).
// ---------------------------------------------------------------------------
__global__ __launch_bounds__(256)
void rnorm_kernel(const float* __restrict__ q, const float* __restrict__ y,
                  float* __restrict__ rn) {
  const int row  = blockIdx.x * 8 + (threadIdx.x >> 5);
  const int lane = threadIdx.x & 31;
  const float* src = (row < BATCH * LSEQ)
                       ? (q + (size_t)row * DDIM)
                       : (y + (size_t)(row - BATCH * LSEQ) * DDIM);
  v2f x = *(const v2f*)(src + 2 * lane);
  float s = x.x * x.x + x.y * x.y;
  s = s + __shfl_xor(s, 16, 32);
  s = s + __shfl_xor(s,  8, 32);
  s = s + __shfl_xor(s,  4, 32);
  s = s + __shfl_xor(s,  2, 32);
  s = s + __shfl_xor(s,  1, 32);
  if (lane == 0) rn[row] = 1.0f / fmaxf(sqrtf(s), EPSN);
}

// ---------------------------------------------------------------------------
// Kernel 2: init sim[] to -inf (harness poisons d_out; atomics need identity).
// ---------------------------------------------------------------------------
__global__ void init_sim_kernel(unsigned int* __restrict__ sim) {
  int i = blockIdx.x * blockDim.x + threadIdx.x;
  if (i < BATCH * LSEQ) sim[i] = 0xFF800000u;  // -inf
}

// Order-independent float atomic max (deterministic across replays).
__device__ __forceinline__ void atomicMaxF(float* addr, float val) {
  if (val >= 0.0f) atomicMax((int*)addr, __float_as_int(val));
  else             atomicMin((unsigned int*)addr, __float_as_uint(val));
}

// ---------------------------------------------------------------------------
// Kernel 3: fused normalized GEMM + row-max.
// Block = 8 waves; wave w owns q-tile rows [q0, q0+16) and sweeps a 256-wide
// k strip (16 tiles of 16). Full fp32 via V_WMMA_F32_16X16X4_F32, D=64 -> 16
// WMMAs per tile. Normalization folded in as per-lane scalar scales.
// ---------------------------------------------------------------------------
__global__ __launch_bounds__(256)
void attgemm_kernel(const float* __restrict__ q, const float* __restrict__ y,
                    const float* __restrict__ rn,
                    float* __restrict__ att, float* __restrict__ sim) {
  const int wave = threadIdx.x >> 5;
  const int lane = threadIdx.x & 31;
  const int lr   = lane & 15;   // row-within-16 (M for A, N for B)
  const int half = lane >> 4;   // K-half selector

  const int batch  = blockIdx.z;
  const int q0     = blockIdx.y * 128 + wave * 16;
  const int n0base = blockIdx.x * 256;

  // ---- A fragments: qrow = row (q0+lr), pre-scaled by 1/||q|| -------------
  const float* qrow = q + ((size_t)batch * LSEQ + q0 + lr) * DDIM;
  const float  rqv  = rn[batch * LSEQ + q0 + lr];
  v2f afr[16];
#pragma unroll
  for (int s = 0; s < 16; ++s) {
    v2f a = *(const v2f*)(qrow + 4 * s + 2 * half);  // K = 4s+2h, 4s+2h+1
    afr[s].x = a.x * rqv;
    afr[s].y = a.y * rqv;
  }

  float rowmax[8];
#pragma unroll
  for (int v = 0; v < 8; ++v) rowmax[v] = -__builtin_inff();

  for (int kt = 0; kt < 16; ++kt) {
    const int n0 = n0base + kt * 16;
    const float* yrow = y + ((size_t)batch * LSEQ + n0 + lr) * DDIM;
    const float  ryv  = rn[BATCH * LSEQ + batch * LSEQ + n0 + lr];

    v8f acc = {};
#pragma unroll
    for (int s = 0; s < 16; ++s) {
      v2f b = *(const v2f*)(yrow + 4 * s + 2 * half);
      b.x *= ryv;
      b.y *= ryv;
      acc = __builtin_amdgcn_wmma_f32_16x16x4_f32(
          /*neg_a=*/false, afr[s], /*neg_b=*/false, b,
          /*c_mod=*/(short)0, acc, /*reuse_a=*/false, /*reuse_b=*/false);
    }

    // D layout: VGPR v -> row (v + 8*half), col lr.
    float* out = att + ((size_t)batch * LSEQ + q0 + 8 * half) * LSEQ + n0 + lr;
#pragma unroll
    for (int v = 0; v < 8; ++v) {
      out[(size_t)v * LSEQ] = acc[v];
      rowmax[v] = fmaxf(rowmax[v], acc[v]);
    }
  }

  // Max over the 16 lanes of each half-wave (columns of this strip).
#pragma unroll
  for (int v = 0; v < 8; ++v) {
    float m = rowmax[v];
    m = fmaxf(m, __shfl_xor(m, 1, 32));
    m = fmaxf(m, __shfl_xor(m, 2, 32));
    m = fmaxf(m, __shfl_xor(m, 4, 32));
    m = fmaxf(m, __shfl_xor(m, 8, 32));
    if (lr == 0)
      atomicMaxF(&sim[batch * LSEQ + q0 + 8 * half + v], m);
  }
}

// ---------------------------------------------------------------------------
extern "C" void kernel_launch(void* const* d_in, const int* in_sizes, int n_in,
                              void* d_out, int out_size, void* d_ws, size_t ws_size,
                              hipStream_t stream) {
  (void)in_sizes; (void)n_in; (void)out_size; (void)ws_size;

  const float* q = (const float*)d_in[0];
  const float* y = (const float*)d_in[1];
  float* att = (float*)d_out;
  float* sim = att + (size_t)BATCH * LSEQ * LSEQ;  // outputs concatenated flat
  float* rn  = (float*)d_ws;                       // 2*B*L floats = 128 KB

  // 32768 rows, 8 rows per 256-thread block.
  rnorm_kernel<<<(2 * BATCH * LSEQ) / 8, 256, 0, stream>>>(q, y, rn);
  init_sim_kernel<<<(BATCH * LSEQ + 255) / 256, 256, 0, stream>>>((unsigned int*)sim);
  // grid: x = 8 k-strips, y = 16 q-blocks (128 rows), z = 8 batches.
  attgemm_kernel<<<dim3(8, 16, 8), 256, 0, stream>>>(q, y, rn, att, sim);
}